// SIRModelRK_75900662055361
// MI455X (gfx1250) — compile-verified
//
#include <hip/hip_runtime.h>
#include <hip/hip_bf16.h>

#define NT   50
#define TPB  64
#define ROW  (NT * 3)          // 150 floats per batch element

__device__ __forceinline__ float fast_div(float a, float b) {
    // v_rcp_f32 (~1 ulp) + one Newton-Raphson refinement -> ~fp32-accurate quotient
    float r = __builtin_amdgcn_rcpf(b);
    r = r * (2.0f - b * r);
    return a * r;
}

__device__ __forceinline__ void sir_eq(float S, float I, float R,
                                       float beta, float gamma,
                                       float& oS, float& oI, float& oR) {
    float n   = (S + I) + R;
    float inf = fast_div(beta * S * I, n);
    float rec = gamma * I;
    oS = S - inf;
    oI = (I + inf) - rec;
    oR = R + rec;
}

__global__ __launch_bounds__(TPB)
void sir_rk4_traj(const float* __restrict__ x,
                  const float* __restrict__ beta_p,
                  const float* __restrict__ gamma_p,
                  float* __restrict__ out,
                  int batch) {
    __shared__ float traj[TPB * ROW];          // 38400 bytes of LDS
    const int   tid   = threadIdx.x;
    const int   b0    = blockIdx.x * TPB;
    const int   b     = b0 + tid;
    const float beta  = beta_p[0];
    const float gamma = gamma_p[0];
    const float H     = 0.1f;                  // STEP
    const float H2    = 0.05f;                 // STEP*k/2 == (STEP/2)*k exactly
    const float C6    = (float)(0.1 / 6.0);    // Python double STEP/6 rounded to f32

    if (b < batch) {
        const float xS = x[3 * b + 0];
        const float xI = x[3 * b + 1];
        const float xR = x[3 * b + 2];
        float* row = &traj[tid * ROW];
        row[0] = xS; row[1] = xI; row[2] = xR;  // timestep 0 = original x
        float pS = xS, pI = xI, pR = xR;
#pragma unroll 1
        for (int t = 1; t < NT; ++t) {
            float k1S,k1I,k1R, k2S,k2I,k2R, k3S,k3I,k3R, k4S,k4I,k4R;
            sir_eq(pS, pI, pR, beta, gamma, k1S, k1I, k1R);
            sir_eq(pS + H2 * k1S, pI + H2 * k1I, pR + H2 * k1R, beta, gamma, k2S, k2I, k2R);
            sir_eq(pS + H2 * k2S, pI + H2 * k2I, pR + H2 * k2R, beta, gamma, k3S, k3I, k3R);
            sir_eq(pS + H  * k3S, pI + H  * k3I, pR + H  * k3R, beta, gamma, k4S, k4I, k4R);
            // NOTE: reference adds the increment to the ORIGINAL x every step.
            float nS = xS + C6 * (k1S + 2.0f * (k2S + k3S) + k4S);
            float nI = xI + C6 * (k1I + 2.0f * (k2I + k3I) + k4I);
            float nR = xR + C6 * (k1R + 2.0f * (k2R + k3R) + k4R);
            row[3 * t + 0] = nS;
            row[3 * t + 1] = nI;
            row[3 * t + 2] = nR;
            pS = nS; pI = nI; pR = nR;
        }
    }
    __syncthreads();   // all LDS trajectory rows visible to the whole workgroup

    // Bulk LDS -> global copy via CDNA5 async stores (ASYNCcnt path).
    // Block's output region is contiguous: out[b0*150 .. (b0+valid)*150).
    const int                valid   = min(TPB, batch - b0);
    const unsigned           bytes   = (unsigned)valid * (unsigned)(ROW * 4);
    const unsigned           nchunks = bytes >> 4;                       // 16B chunks
    const unsigned long long gbase   = (unsigned long long)(out + (size_t)b0 * ROW);
    const unsigned           lbase   = (unsigned)(size_t)(&traj[0]);     // LDS offset = low 32b of flat addr

    for (unsigned i = (unsigned)tid; i < nchunks; i += TPB) {
        unsigned long long ga = gbase + ((unsigned long long)i << 4);
        unsigned           la = lbase + (i << 4);
        asm volatile("global_store_async_from_lds_b128 %0, %1, off"
                     :: "v"(ga), "v"(la) : "memory");
    }
    if ((bytes & 15u) != 0u && tid == 0) {     // 8B remainder (only if valid is odd)
        unsigned long long ga = gbase + ((unsigned long long)nchunks << 4);
        unsigned           la = lbase + (nchunks << 4);
        asm volatile("global_store_async_from_lds_b64 %0, %1, off"
                     :: "v"(ga), "v"(la) : "memory");
    }
    asm volatile("s_wait_asynccnt 0" ::: "memory");
}

extern "C" void kernel_launch(void* const* d_in, const int* in_sizes, int n_in,
                              void* d_out, int out_size, void* d_ws, size_t ws_size,
                              hipStream_t stream) {
    const float* x     = (const float*)d_in[0];
    const float* beta  = (const float*)d_in[1];
    const float* gamma = (const float*)d_in[2];
    float*       out   = (float*)d_out;

    const int batch  = in_sizes[0] / 3;
    const int blocks = (batch + TPB - 1) / TPB;
    sir_rk4_traj<<<blocks, TPB, 0, stream>>>(x, beta, gamma, out, batch);
}